// SmallVDSR_F8_1580547973402
// MI455X (gfx1250) — compile-verified
//
#include <hip/hip_runtime.h>

// ---------------- types ----------------
typedef float v2f __attribute__((ext_vector_type(2)));
typedef float v8f __attribute__((ext_vector_type(8)));

#define IMG_H 512
#define IMG_W 512
#define NCH   8
// mid-layer tile: 64 wide x 16 tall output, 1-px halo staged in LDS
#define TW     64
#define TH     16
#define LW     66
#define LH     18
#define LPITCH 68
#define CH_STRIDE (LH * LPITCH)

// ---------------- conv1: 1 -> 8, relu (memory bound, direct) ----------------
__global__ __launch_bounds__(256) void conv1_1to8_relu(
    const float* __restrict__ in,   // [8][1][512][512]
    const float* __restrict__ w,    // [8][1][3][3]
    float* __restrict__ out) {      // [8][8][512][512]
  int idx = blockIdx.x * 256 + threadIdx.x;       // one thread per pixel
  int n   = idx >> 18;
  int rem = idx & ((1 << 18) - 1);
  int y   = rem >> 9;
  int x   = rem & 511;
  float v[3][3];
#pragma unroll
  for (int ky = 0; ky < 3; ++ky)
#pragma unroll
    for (int kx = 0; kx < 3; ++kx) {
      int gy = y + ky - 1, gx = x + kx - 1;
      v[ky][kx] = ((unsigned)gy < IMG_H && (unsigned)gx < IMG_W)
                      ? in[(n << 18) + (gy << 9) + gx] : 0.f;
    }
#pragma unroll
  for (int o = 0; o < 8; ++o) {
    float s = 0.f;
#pragma unroll
    for (int t = 0; t < 9; ++t) s += w[o * 9 + t] * v[t / 3][t % 3];
    out[((n * NCH + o) << 18) + (y << 9) + x] = fmaxf(s, 0.f);
  }
}

// ---------------- mid layers: 8 -> 8, relu, fp32 WMMA ----------------
// D = A x B + C with V_WMMA_F32_16X16X4_F32:
//   A (16x4)  = weights: M = out channel (8 used), K = 4 input channels
//   B (4x16)  = activations: K = 4 input channels, N = 16 pixels
//   9 taps * 2 K-chunks = 18 WMMAs per 16-pixel tile.
__global__ __launch_bounds__(256) void conv_mid_8to8_wmma(
    const float* __restrict__ in,   // [8][8][512][512]
    const float* __restrict__ w,    // [8][8][3][3]
    float* __restrict__ out,        // [8][8][512][512]
    int relu) {
  __shared__ float lds[NCH * LH * LPITCH];

  const int tid = threadIdx.x;
  const int n   = blockIdx.z;
  const int ty0 = blockIdx.y * TH;
  const int tx0 = blockIdx.x * TW;

  // ---- stage 8ch x 18 x 66 halo tile into LDS (zero-padded borders) ----
  for (int i = tid; i < NCH * LH * LW; i += 256) {
    int col = i % LW;
    int t   = i / LW;
    int row = t % LH;
    int c   = t / LH;
    int gy = ty0 + row - 1, gx = tx0 + col - 1;
    float v = 0.f;
    if ((unsigned)gy < IMG_H && (unsigned)gx < IMG_W)
      v = in[((n * NCH + c) << 18) + (gy << 9) + gx];
    lds[(c * LH + row) * LPITCH + col] = v;
  }

  // ---- preload 18 A-fragments (weights), reused for all tiles ----
  const int lane = tid & 31;
  const int wave = tid >> 5;
  const int o    = lane & 15;   // output channel row (A: M across lanes 0..15)
  const int hi   = lane >> 4;   // 0 -> K=0,1 ; 1 -> K=2,3 (f32 A 16x4 layout)
  v2f A[18];
#pragma unroll
  for (int t = 0; t < 9; ++t) {
#pragma unroll
    for (int ch = 0; ch < 2; ++ch) {
      int c0 = ch * 4 + hi * 2;
      float a0 = 0.f, a1 = 0.f;
      if (o < 8) {
        a0 = w[(o * 8 + c0) * 9 + t];
        a1 = w[(o * 8 + c0 + 1) * 9 + t];
      }
      v2f a; a.x = a0; a.y = a1;
      A[t * 2 + ch] = a;
    }
  }
  __syncthreads();

  const int px  = lane & 15;  // pixel within tile (B: N across lanes 0..15)
  const int hi2 = hi * 2;     // upper half-wave supplies K=2,3 rows of B

  // 64 tile-groups (16 rows x 4 col-groups) over 8 waves -> 8 each
  for (int g = wave; g < 64; g += 8) {
    int r  = g & 15;          // row within tile
    int tx = (g >> 4) << 4;   // 16-pixel column group

    v8f acc = {0.f, 0.f, 0.f, 0.f, 0.f, 0.f, 0.f, 0.f};
#pragma unroll
    for (int t = 0; t < 9; ++t) {
      int ky = t / 3, kx = t % 3;
#pragma unroll
      for (int ch = 0; ch < 2; ++ch) {
        int c = ch * 4 + hi2;
        const float* p = &lds[(c * LH + r + ky) * LPITCH + tx + px + kx];
        v2f b; b.x = p[0]; b.y = p[CH_STRIDE];   // channels c, c+1
        acc = __builtin_amdgcn_wmma_f32_16x16x4_f32(
            false, A[t * 2 + ch], false, b, (short)0, acc, false, false);
      }
    }

    // D layout: VGPR rr, lanes 0..15 -> out channel rr, pixel = lane
    if (hi == 0) {
      int gy = ty0 + r;
      int gx = tx0 + tx + px;
      float* op = &out[((n * NCH) << 18) + (gy << 9) + gx];
#pragma unroll
      for (int rr = 0; rr < 8; ++rr) {
        float vv = acc[rr];
        if (relu) vv = fmaxf(vv, 0.f);
        op[rr << 18] = vv;
      }
    }
  }
}

// ---------------- conv20: 8 -> 1, no relu (LDS-tiled direct) ----------------
__global__ __launch_bounds__(256) void conv20_8to1(
    const float* __restrict__ in,   // [8][8][512][512]
    const float* __restrict__ w,    // [1][8][3][3]
    float* __restrict__ out) {      // [8][1][512][512]
  __shared__ float lds[NCH * LH * LPITCH];

  const int tid = threadIdx.x;
  const int n   = blockIdx.z;
  const int ty0 = blockIdx.y * TH;
  const int tx0 = blockIdx.x * TW;

  for (int i = tid; i < NCH * LH * LW; i += 256) {
    int col = i % LW;
    int t   = i / LW;
    int row = t % LH;
    int c   = t / LH;
    int gy = ty0 + row - 1, gx = tx0 + col - 1;
    float v = 0.f;
    if ((unsigned)gy < IMG_H && (unsigned)gx < IMG_W)
      v = in[((n * NCH + c) << 18) + (gy << 9) + gx];
    lds[(c * LH + row) * LPITCH + col] = v;
  }
  __syncthreads();

  for (int p = tid; p < TW * TH; p += 256) {
    int lx = p & 63;
    int ly = p >> 6;
    float s = 0.f;
#pragma unroll
    for (int c = 0; c < 8; ++c)
#pragma unroll
      for (int t = 0; t < 9; ++t)
        s += w[c * 9 + t] * lds[(c * LH + ly + t / 3) * LPITCH + lx + t % 3];
    out[(n << 18) + ((ty0 + ly) << 9) + (tx0 + lx)] = s;
  }
}

// ---------------- launcher ----------------
extern "C" void kernel_launch(void* const* d_in, const int* in_sizes, int n_in,
                              void* d_out, int out_size, void* d_ws, size_t ws_size,
                              hipStream_t stream) {
  const float* x    = (const float*)d_in[0];  // [8][1][512][512]
  const float* w1   = (const float*)d_in[1];  // [8][1][3][3]
  const float* wmid = (const float*)d_in[2];  // [18][8][8][3][3]
  const float* w20  = (const float*)d_in[3];  // [1][8][3][3]
  float* out = (float*)d_out;

  // ping-pong feature maps: 2 x 8*8*512*512 floats = 2 x 64 MB in d_ws
  const size_t FEAT = (size_t)8 * NCH * IMG_H * IMG_W;  // elements
  float* bufA = (float*)d_ws;
  float* bufB = bufA + FEAT;
  (void)ws_size; (void)in_sizes; (void)n_in; (void)out_size;

  dim3 gridT(IMG_W / TW, IMG_H / TH, 8);  // (8, 32, 8)

  // conv1 + relu
  conv1_1to8_relu<<<(8 * IMG_H * IMG_W) / 256, 256, 0, stream>>>(x, w1, bufA);

  // conv2..conv19: 18 x (8->8 conv + relu), ping-pong
  float* src = bufA;
  float* dst = bufB;
  for (int L = 0; L < 18; ++L) {
    conv_mid_8to8_wmma<<<gridT, 256, 0, stream>>>(src, wmid + (size_t)L * 8 * 8 * 9,
                                                  dst, /*relu=*/1);
    float* t = src; src = dst; dst = t;
  }

  // conv20, no relu
  conv20_8to1<<<gridT, 256, 0, stream>>>(src, w20, out);
}